// IDVT_Encoder_26173530702193
// MI455X (gfx1250) — compile-verified
//
#include <hip/hip_runtime.h>

#define DD 64

typedef __attribute__((ext_vector_type(2))) float v2f;
typedef __attribute__((ext_vector_type(8))) float v8f;

__device__ __forceinline__ void atomAddF(float* p, float v) {
    __hip_atomic_fetch_add(p, v, __ATOMIC_RELAXED, __HIP_MEMORY_SCOPE_AGENT);
}

// ---------------------------------------------------------------------------
// SpMM: y[row[e], :] += val[e] * x[col[e], :]    (wave per edge, lane = 2 cols)
// ---------------------------------------------------------------------------
__global__ void k_spmm(const int* __restrict__ row, const int* __restrict__ col,
                       const float* __restrict__ val, const float* __restrict__ x,
                       float* __restrict__ y, int nedges) {
    long long gid = (long long)blockIdx.x * blockDim.x + threadIdx.x;
    int e = (int)(gid >> 5);
    int lane = threadIdx.x & 31;
    if (e >= nedges) return;
    int r = row[e], c = col[e];
    float v = val[e];
    v2f xv = *(const v2f*)(x + (size_t)c * DD + 2 * lane);
    float* yp = y + (size_t)r * DD + 2 * lane;
    atomAddF(yp + 0, v * xv.x);
    atomAddF(yp + 1, v * xv.y);
}

// ---------------------------------------------------------------------------
// Row L2-normalize in place (wave per row), denom = max(norm, 1e-8)
// ---------------------------------------------------------------------------
__global__ void k_rownorm(float* __restrict__ u, int nrows) {
    long long gid = (long long)blockIdx.x * blockDim.x + threadIdx.x;
    int r = (int)(gid >> 5);
    int lane = threadIdx.x & 31;
    if (r >= nrows) return;
    v2f v = *(v2f*)(u + (size_t)r * DD + 2 * lane);
    float p = v.x * v.x + v.y * v.y;
    for (int m = 16; m > 0; m >>= 1) p += __shfl_xor(p, m, 32);
    float inv = 1.0f / fmaxf(sqrtf(p), 1e-8f);
    v.x *= inv; v.y *= inv;
    *(v2f*)(u + (size_t)r * DD + 2 * lane) = v;
}

// ---------------------------------------------------------------------------
// Per-edge cosine sim -> sim_value; block-reduced partial sum for the mean
// ---------------------------------------------------------------------------
__global__ void k_sims(const int* __restrict__ row, const int* __restrict__ col,
                       const float* __restrict__ un, float* __restrict__ sval,
                       float* __restrict__ mean_acc, int nedges) {
    long long gid = (long long)blockIdx.x * blockDim.x + threadIdx.x;
    int e = (int)(gid >> 5);
    int lane = threadIdx.x & 31;
    float sv = 0.0f;
    if (e < nedges) {
        int r = row[e], c = col[e];
        v2f a = *(const v2f*)(un + (size_t)r * DD + 2 * lane);
        v2f b = *(const v2f*)(un + (size_t)c * DD + 2 * lane);
        float p = a.x * b.x + a.y * b.y;
        for (int m = 16; m > 0; m >>= 1) p += __shfl_xor(p, m, 32);
        sv = (p + 1.0f) * 0.5f;
        if (lane == 0) sval[e] = sv;
    }
    __shared__ float ssum[8];
    int wv = threadIdx.x >> 5;
    if (lane == 0) ssum[wv] = (e < nedges) ? sv : 0.0f;
    __syncthreads();
    if (threadIdx.x == 0) {
        float s = 0.0f;
        for (int i = 0; i < 8; i++) s += ssum[i];
        atomAddF(mean_acc, s);
    }
}

__global__ void k_thresh(float* scal, int nedges) {
    scal[1] = (scal[0] / (float)nedges > 0.7f) ? 0.8f : 0.0f;
}

__global__ void k_prune(float* __restrict__ sval, const int* __restrict__ row,
                        float* __restrict__ diags, const float* __restrict__ scal,
                        int nedges) {
    int e = blockIdx.x * blockDim.x + threadIdx.x;
    if (e >= nedges) return;
    float thr = scal[1];
    float s = sval[e];
    s = (s < thr) ? 0.0f : s;
    sval[e] = s;
    if (s != 0.0f) atomAddF(&diags[row[e]], s);
}

__global__ void k_edgenorm(float* __restrict__ sval, const int* __restrict__ row,
                           const float* __restrict__ diags, int nedges) {
    int e = blockIdx.x * blockDim.x + threadIdx.x;
    if (e >= nedges) return;
    sval[e] = sval[e] / (diags[row[e]] + 1e-7f);
}

__global__ void k_axpy(float* __restrict__ dst, const float* __restrict__ src, int n) {
    int i = blockIdx.x * blockDim.x + threadIdx.x;
    if (i < n) dst[i] += src[i];
}

__global__ void k_scale(float* __restrict__ dst, float s, int n) {
    int i = blockIdx.x * blockDim.x + threadIdx.x;
    if (i < n) dst[i] *= s;
}

__global__ void k_egoinit(const float* __restrict__ ue, const float* __restrict__ ie,
                          const float* __restrict__ sv, float* __restrict__ ego,
                          float* __restrict__ out, int nud, int tot) {
    int i = blockIdx.x * blockDim.x + threadIdx.x;
    if (i >= tot) return;
    float v = (i < nud) ? (ue[i] + sv[i]) : ie[i - nud];
    ego[i] = v;
    out[i] = v;
}

// ---------------------------------------------------------------------------
// Gate: out[:NU] = gu*sview + (1-gu)*user_v1,
//       gu = sigmoid(user_v1 @ w1^T + sview @ w2^T), fp32 WMMA 16x16x4.
// Wave per 16x16 tile. A (16x4): lane m=lane&15, VGPR0/1 = K{2h, 2h+1}.
// B (4x16): lane n=lane&15, VGPR0/1 = K{2h, 2h+1}, B[k][n] = w[n][k].
// C/D: vgpr r -> M = r + 8*(lane>>4), N = lane&15.
// ---------------------------------------------------------------------------
__global__ void k_gate(float* __restrict__ out, const float* __restrict__ sview,
                       const float* __restrict__ w1, const float* __restrict__ w2,
                       int ntm) {
    long long gid = (long long)blockIdx.x * blockDim.x + threadIdx.x;
    int wave = (int)(gid >> 5);
    int lane = threadIdx.x & 31;
    int mt = wave >> 2;          // DD/16 = 4 n-tiles
    int nt = wave & 3;
    if (mt >= ntm) return;       // uniform per wave -> EXEC all-ones inside
    const int m0 = mt * 16, n0 = nt * 16;
    const int mloc = lane & 15;
    const int h = lane >> 4;
    const float* A1 = out + (size_t)(m0 + mloc) * DD;     // user_v1 rows
    const float* A2 = sview + (size_t)(m0 + mloc) * DD;   // sview rows
    const float* B1 = w1 + (size_t)(n0 + mloc) * DD;      // w^T columns
    const float* B2 = w2 + (size_t)(n0 + mloc) * DD;
    v8f c = {};
#pragma unroll
    for (int k = 0; k < DD; k += 4) {
        v2f a = *(const v2f*)(A1 + k + 2 * h);
        v2f b = *(const v2f*)(B1 + k + 2 * h);
        c = __builtin_amdgcn_wmma_f32_16x16x4_f32(false, a, false, b,
                                                  (short)0, c, false, false);
    }
#pragma unroll
    for (int k = 0; k < DD; k += 4) {
        v2f a = *(const v2f*)(A2 + k + 2 * h);
        v2f b = *(const v2f*)(B2 + k + 2 * h);
        c = __builtin_amdgcn_wmma_f32_16x16x4_f32(false, a, false, b,
                                                  (short)0, c, false, false);
    }
    const int col = n0 + mloc;
#pragma unroll
    for (int r = 0; r < 8; r++) {
        int g = m0 + r + 8 * h;
        float gu = 1.0f / (1.0f + __expf(-c[r]));
        size_t idx = (size_t)g * DD + col;
        float v1 = out[idx];
        float sv = sview[idx];
        out[idx] = gu * sv + (1.0f - gu) * v1;
    }
}

// ---------------------------------------------------------------------------
extern "C" void kernel_launch(void* const* d_in, const int* in_sizes, int n_in,
                              void* d_out, int out_size, void* d_ws, size_t ws_size,
                              hipStream_t stream) {
    (void)n_in; (void)out_size; (void)ws_size;
    const float* user_emb = (const float*)d_in[0];
    const float* item_emb = (const float*)d_in[1];
    const float* w1 = (const float*)d_in[2];
    const float* w2 = (const float*)d_in[3];
    const int* soc_row = (const int*)d_in[4];
    const int* soc_col = (const int*)d_in[5];
    const float* soc_val = (const float*)d_in[6];
    const int* adj_row = (const int*)d_in[7];
    const int* adj_col = (const int*)d_in[8];
    const float* adj_val = (const float*)d_in[9];

    const int nu = in_sizes[0] / DD;
    const int ni = in_sizes[1] / DD;
    const int e_soc = in_sizes[4];
    const int e_ui = in_sizes[7];
    const size_t NUD = (size_t)nu * DD;
    const size_t TOT = (size_t)(nu + ni) * DD;

    float* outp = (float*)d_out;

    // Workspace layout: persistent sview, then an arena reused across phases.
    float* sview = (float*)d_ws;
    float* arena = sview + NUD;
    // early phase (social graph):
    float* u     = arena;                 // spmm result / normalized / 3rd ping-pong
    float* ue0   = arena + NUD;
    float* ue1   = arena + 2 * NUD;
    float* sval  = arena + 3 * NUD;
    float* diags = sval + e_soc;
    float* scal  = diags + nu;            // [0]=mean accum, [1]=threshold
    // ego phase overlays the whole arena (early buffers all dead by then):
    float* ego0 = arena;
    float* ego1 = arena + TOT;

    const int B = 256;
    long long t;

    // ---- similarity adjacency ----
    hipMemsetAsync(u, 0, NUD * sizeof(float), stream);
    t = (long long)e_soc * 32;
    k_spmm<<<(t + B - 1) / B, B, 0, stream>>>(soc_row, soc_col, soc_val, user_emb, u, e_soc);
    t = (long long)nu * 32;
    k_rownorm<<<(t + B - 1) / B, B, 0, stream>>>(u, nu);
    hipMemsetAsync(scal, 0, 2 * sizeof(float), stream);
    t = (long long)e_soc * 32;
    k_sims<<<(t + B - 1) / B, B, 0, stream>>>(soc_row, soc_col, u, sval, scal, e_soc);
    k_thresh<<<1, 1, 0, stream>>>(scal, e_soc);
    hipMemsetAsync(diags, 0, (size_t)nu * sizeof(float), stream);
    k_prune<<<(e_soc + B - 1) / B, B, 0, stream>>>(sval, soc_row, diags, scal, e_soc);
    k_edgenorm<<<(e_soc + B - 1) / B, B, 0, stream>>>(sval, soc_row, diags, e_soc);

    // ---- 3 social propagation layers, accumulate into sview ----
    hipMemsetAsync(sview, 0, NUD * sizeof(float), stream);
    const float* src = user_emb;
    float* dsts[3] = {ue0, ue1, u};
    for (int l = 0; l < 3; l++) {
        float* dst = dsts[l];
        hipMemsetAsync(dst, 0, NUD * sizeof(float), stream);
        t = (long long)e_soc * 32;
        k_spmm<<<(t + B - 1) / B, B, 0, stream>>>(soc_row, soc_col, sval, src, dst, e_soc);
        k_axpy<<<((int)NUD + B - 1) / B, B, 0, stream>>>(sview, dst, (int)NUD);
        src = dst;
    }
    k_scale<<<((int)NUD + B - 1) / B, B, 0, stream>>>(sview, 1.0f / 3.0f, (int)NUD);

    // ---- 2 user-item layers, accumulate directly into d_out ----
    k_egoinit<<<((int)TOT + B - 1) / B, B, 0, stream>>>(user_emb, item_emb, sview,
                                                        ego0, outp, (int)NUD, (int)TOT);
    float* cur = ego0;
    float* nxt = ego1;
    for (int l = 0; l < 2; l++) {
        hipMemsetAsync(nxt, 0, TOT * sizeof(float), stream);
        t = (long long)e_ui * 32;
        k_spmm<<<(t + B - 1) / B, B, 0, stream>>>(adj_row, adj_col, adj_val, cur, nxt, e_ui);
        k_axpy<<<((int)TOT + B - 1) / B, B, 0, stream>>>(outp, nxt, (int)TOT);
        float* tmp = cur; cur = nxt; nxt = tmp;
    }
    k_scale<<<((int)TOT + B - 1) / B, B, 0, stream>>>(outp, 1.0f / 3.0f, (int)TOT);

    // ---- fused WMMA gate (in place on user rows of d_out) ----
    int ntm = nu / 16;
    t = (long long)ntm * (DD / 16) * 32;
    k_gate<<<(t + B - 1) / B, B, 0, stream>>>(outp, sview, w1, w2, ntm);
}